// RoIAlignMax_64974265254147
// MI455X (gfx1250) — compile-verified
//
#include <hip/hip_runtime.h>

// ---------------------------------------------------------------------------
// RoIAlign (legacy 8x8 single-sample grid) + 2x2/s1 max-pool for MI455X.
// Bandwidth-bound (~0.3 GFLOP vs ~320MB HBM traffic): transpose features to
// channel-last once (stays resident in the 192MB L2; filled via gfx1250
// global_load_async_to_lds_b128), then gathers are contiguous channel rows and
// the bilinear blend runs on the WMMA unit (v_wmma_f32_16x16x4_f32).
// ---------------------------------------------------------------------------

typedef __attribute__((ext_vector_type(2))) float    v2f;
typedef __attribute__((ext_vector_type(4))) float    v4f;
typedef __attribute__((ext_vector_type(8))) float    v8f;
typedef __attribute__((ext_vector_type(2))) unsigned v2u;

#define RA_B 2
#define RA_C 256
#define RA_H 200
#define RA_W 272
#define RA_HW (RA_H * RA_W)    // 54400
#define GP 64                  // 8x8 grid points per ROI
#define PITCH 65               // LDS pitch (64 gps + 1) -> conflict-free
#define HALF_C 128             // channels per LDS pass
#define T_HW 128               // transpose tile width (54400 = 425*128)
#define T_PITCH (T_HW + 4)     // LDS pitch for transpose tile

// --------------------------- transpose kernel ------------------------------
// [B,C,H*W] -> [B,H*W,C].  Source rows are pulled into LDS with the CDNA5
// async memory->LDS DMA path (ASYNCcnt), then written back channel-last as
// coalesced float4 stores (these land in L2 and get reused by the gathers).
__global__ __launch_bounds__(256) void transpose_nchw_to_nhwc(
    const float* __restrict__ in, float* __restrict__ out)
{
    __shared__ float tile[32][T_PITCH];    // 32 channels x 128 hw, padded
    const int b   = blockIdx.z;
    const int hw0 = blockIdx.x * T_HW;     // 425 blocks
    const int c0  = blockIdx.y * 32;       // 8 blocks
    const int tx  = threadIdx.x & 31;      // hw float4 index (0..31)
    const int ty  = threadIdx.x >> 5;      // c row (0..7)

#if defined(__gfx1250__)
#pragma unroll
    for (int i = 0; i < 4; ++i) {
        const int cl = ty + i * 8;
        const float* src =
            &in[((size_t)b * RA_C + c0 + cl) * RA_HW + hw0 + tx * 4];
        // Generic LDS pointer: low 32 bits are the LDS byte offset.
        const unsigned lds_off =
            (unsigned)(unsigned long long)(&tile[cl][tx * 4]);
        const unsigned long long gaddr = (unsigned long long)src;
        asm volatile("global_load_async_to_lds_b128 %0, %1, off"
                     :: "v"(lds_off), "v"(gaddr) : "memory");
    }
    asm volatile("s_wait_asynccnt 0x0" ::: "memory");
#else
#pragma unroll
    for (int i = 0; i < 4; ++i) {
        const int cl = ty + i * 8;
        const v4f v = __builtin_nontemporal_load((const v4f*)
            &in[((size_t)b * RA_C + c0 + cl) * RA_HW + hw0 + tx * 4]);
        *(v4f*)&tile[cl][tx * 4] = v;
    }
#endif
    __syncthreads();

    // Write channel-last: 8 c-groups x 32 hw rows per slab, 4 slabs.
    const int cg  = threadIdx.x & 7;       // float4 group along c
    const int hwl = threadIdx.x >> 3;      // 0..31
#pragma unroll
    for (int s = 0; s < 4; ++s) {
        const int hw = hwl + s * 32;
        v4f v;
        v.x = tile[cg * 4 + 0][hw];
        v.y = tile[cg * 4 + 1][hw];
        v.z = tile[cg * 4 + 2][hw];
        v.w = tile[cg * 4 + 3][hw];
        *(v4f*)&out[((size_t)b * RA_HW + hw0 + hw) * RA_C + c0 + cg * 4] = v;
    }
}

// ------------------------ fused align + maxpool ----------------------------
// One block (256 threads = 8 wave32 waves) per ROI.  Compile-time strides
// (elements): channel-last fast path or NCHW fallback.  All gather addressing
// is 32-bit (tensor < 2^31 elements).
template <int CH_S, int X_S, int Y_S, int B_S>
__global__ __launch_bounds__(256) void roialign_maxpool(
    const float* __restrict__ feat, const float* __restrict__ rois,
    float* __restrict__ out)
{
    __shared__ v4f   s_w4[GP];                      // {w00,w01,w10,w11} per gp
    __shared__ int   s_off[GP];                     // y0*Y_S + x0*X_S per gp
    __shared__ v2u   s_qtbl[49];                    // quad -> 4 packed lidx u16
    __shared__ float s_blend[HALF_C * PITCH + 32];  // + dump slots (hi lanes)

    const int r    = blockIdx.x;
    const int tid  = threadIdx.x;
    const int lane = tid & 31;                      // wave32
    const int wave = tid >> 5;                      // 0..7

    // ROI parameters (block-uniform, recomputed per thread)
    const float bidx = rois[(size_t)r * 5 + 0];
    const float x1   = rois[(size_t)r * 5 + 1] * 0.25f;
    const float y1   = rois[(size_t)r * 5 + 2] * 0.25f;
    const float x2   = rois[(size_t)r * 5 + 3] * 0.25f;
    const float y2   = rois[(size_t)r * 5 + 4] * 0.25f;
    const float binh = fmaxf(y2 - y1, 0.0f) * (1.0f / 7.0f);
    const float binw = fmaxf(x2 - x1, 0.0f) * (1.0f / 7.0f);
    const int   biOff = (int)bidx * B_S;

    // Per-grid-point params.  Validity folded into weights (all-zero weights
    // == reference's post-blend zeroing; pool then sees exact zeros).
    if (tid < GP) {
        const int   ph = tid >> 3, pw = tid & 7;
        const float h  = y1 + (float)ph * binh;
        const float w  = x1 + (float)pw * binw;
        const bool  valid = (h >= 0.0f) && (h < (float)RA_H) &&
                            (w >= 0.0f) && (w < (float)RA_W);
        const float h0 = fminf(fmaxf(floorf(h), 0.0f), (float)(RA_H - 2));
        const float w0 = fminf(fmaxf(floorf(w), 0.0f), (float)(RA_W - 2));
        const float fh = h - h0, fw = w - w0;
        const float m  = valid ? 1.0f : 0.0f;
        v4f wv;
        wv.x = m * (1.0f - fh) * (1.0f - fw);   // (y0  , x0  )
        wv.y = m * (1.0f - fh) * fw;            // (y0  , x0+1)
        wv.z = m * fh * (1.0f - fw);            // (y0+1, x0  )
        wv.w = m * fh * fw;                     // (y0+1, x0+1)
        s_w4[tid]  = wv;
        s_off[tid] = (int)h0 * Y_S + (int)w0 * X_S;
    }
    // Quad table: the (channel-local, grid-point) pattern of 4-output quads
    // repeats every 49 quads (196 outputs == exactly 4 channels).  Entry t
    // packs the 4 LDS indices (c_local*PITCH + ph*8 + pw) as u16.
    if (tid < 49) {
        unsigned lo = 0, hi = 0;
#pragma unroll
        for (int k = 0; k < 4; ++k) {
            const int e   = tid * 4 + k;
            const int cl  = e / 49;
            const int pos = e - cl * 49;
            const int ph  = pos / 7;
            const int gp  = ph * 8 + (pos - ph * 7);
            const unsigned entry = (unsigned)(cl * PITCH + gp);
            if (k < 2) lo |= entry << (16 * k);
            else       hi |= entry << (16 * (k - 2));
        }
        v2u t; t.x = lo; t.y = hi;
        s_qtbl[tid] = t;
    }

    const bool lo16     = (lane < 16);
    const int  wByte    = lo16 ? 0 : 8;         // weight-pair byte offset
    const int  dumpBase = HALF_C * PITCH;

    for (int pass = 0; pass < 2; ++pass) {
        __syncthreads();
        const int cb      = pass * HALF_C;
        const int chLocal = wave * 16 + (lane & 15);          // 0..127
        // Loop-invariant part of the gather offset (32-bit):
        const int chOff   = biOff + (cb + chLocal) * CH_S + (lo16 ? 0 : X_S);

        // ---- stage 1: WMMA bilinear blend, 16 channels per wave per gp ----
        // V_WMMA_F32_16X16X4_F32: D(16x16) = A(16x4) x B(4x16)
        //   A rows identical = this gp's weights (lanes 0-15: K0,K1;
        //   lanes 16-31: K2,K3 per the ISA A layout)
        //   B row-striped: lo lanes carry column x0, hi lanes x0+1;
        //   b0 = row y0, b1 = row y0+1 (inst-offset folds +Y_S).
        //   D row 0 = blended channel vector; compute is free (L2 bound).
#pragma unroll 4
        for (int gp = 0; gp < GP; ++gp) {
            const int off = chOff + s_off[gp];
            v2f bm;
            bm.x = feat[off];
            bm.y = feat[off + Y_S];
            const v2f am = *(const v2f*)((const char*)&s_w4[gp] + wByte);
            v8f acc = {};
            float val;
#if defined(__HIP_DEVICE_COMPILE__)
            v8f d = __builtin_amdgcn_wmma_f32_16x16x4_f32(
                false, am, false, bm, (short)0, acc, false, false);
            val = d[0];
#else
            val = am.x * bm.x + am.y * bm.y;  // host stub only
#endif
            // Branch-free store: hi half-wave writes to dump slots so EXEC
            // stays all-ones around the WMMAs.
            const int sdst = lo16 ? (chLocal * PITCH + gp)
                                  : (dumpBase + (lane - 16));
            s_blend[sdst] = val;
        }
        __syncthreads();

        // ---- stage 2: 2x2/s1 max-pool from LDS, b128 NT stores ------------
        // Output region for this pass is contiguous: 6272 floats = 1568
        // quads.  Per quad: one exact magic div (q/49, valid since 6q < 2^18),
        // one b64 table load, 4 independent pooled maxes, one NT b128 store.
        float* outBase = out + (size_t)r * (RA_C * 49) + (size_t)cb * 49;
        for (int q = tid; q < HALF_C * 49 / 4; q += 256) {
            const int g = (int)(((unsigned)q * 5350u) >> 18);  // q / 49
            const int t = q - g * 49;
            const v2u e2   = s_qtbl[t];
            const int base = g * (4 * PITCH);                  // (4g)*PITCH
            v4f v;
#pragma unroll
            for (int k = 0; k < 4; ++k) {
                const unsigned half = (k < 2) ? e2.x : e2.y;
                const int lidx = base + (int)((half >> (16 * (k & 1))) & 0xFFFFu);
                const float* p = &s_blend[lidx];
                v[k] = fmaxf(fmaxf(p[0], p[1]), fmaxf(p[8], p[9]));
            }
            __builtin_nontemporal_store(v, (v4f*)(outBase + q * 4));
        }
    }
}

// ------------------------------ launcher -----------------------------------
extern "C" void kernel_launch(void* const* d_in, const int* in_sizes, int n_in,
                              void* d_out, int out_size, void* d_ws, size_t ws_size,
                              hipStream_t stream)
{
    const float* feat = (const float*)d_in[0];   // [2,256,200,272] fp32
    const float* rois = (const float*)d_in[1];   // [R,5] fp32
    float* out = (float*)d_out;                  // [R,256,7,7] fp32
    const int R = in_sizes[1] / 5;

    const size_t tbytes = (size_t)RA_B * RA_C * RA_HW * sizeof(float);
    if (ws_size >= tbytes) {
        // Fast path: channel-last staging in ws (L2-resident gathers).
        float* tfeat = (float*)d_ws;
        transpose_nchw_to_nhwc<<<dim3(RA_HW / T_HW, RA_C / 32, RA_B), 256, 0,
                                 stream>>>(feat, tfeat);
        roialign_maxpool<1, RA_C, RA_W * RA_C, RA_HW * RA_C>
            <<<R, 256, 0, stream>>>(tfeat, rois, out);
    } else {
        // Fallback: gather directly from the NCHW layout.
        roialign_maxpool<RA_HW, 1, RA_W, RA_C * RA_HW>
            <<<R, 256, 0, stream>>>(feat, rois, out);
    }
}